// GPUParametricEQ_8323646620202
// MI455X (gfx1250) — compile-verified
//
#include <hip/hip_runtime.h>

// ---------------------------------------------------------------------------
// GPUParametricEQ on MI455X (gfx1250, wave32)
//
// Exact block state-space parallelization of a 10-band biquad cascade:
//   y = F(x, s0) = F(x, 0) + Theta * s0          (linearity, LTI system)
//   s_chunk[c+1] = A^L * s_chunk[c] + E[c]       (chunk-state scan)
//
// Pass 0: coefficients + basis sims -> Theta (L x 20), P = A^L, P32 = A^(32L)
// Pass 1: per-(row,chunk) zero-state recurrence, LDS-staged coalesced I/O
//         (async global->LDS B128 via gfx1250 builtin, ASYNCcnt tracked)
// Pass 2: hierarchical chunk-state scan (depth 32+16+32 instead of 512)
// Pass 3: y += Theta * S  via V_WMMA_F32_16X16X4_F32 (exact f32 correction)
// ---------------------------------------------------------------------------

#define EQ_B   64
#define EQ_T   65536
#define EQ_L   128          // chunk length (time)
#define EQ_NC  512          // EQ_T / EQ_L
#define EQ_NB  10           // bands
#define EQ_NS  20           // state dim (2 per band)
#define EQ_GRP 32           // chunks per scan group
#define EQ_NG  16           // groups per row (EQ_NC / EQ_GRP)
#define EQ_PI  3.14159265358979323846f
#define EQ_FS  16000.0f

// pass-1 blocking
#define P1_THREADS 64       // threads (= chunks) per block
#define P1_STRIDE  132      // padded LDS chunk stride in floats (16B aligned)
#define P1_SEGS    8        // blocks per row (EQ_NC / P1_THREADS)

// workspace layout (in floats)
#define COEFF_OFF  0                                    // 10 x 5 (b0,b1,b2,a1,a2)
#define THETA_OFF  64                                   // EQ_L x EQ_NS
#define P_OFF      (THETA_OFF + EQ_L * EQ_NS)           // EQ_NS x EQ_NS = A^L
#define P32_OFF    (P_OFF + EQ_NS * EQ_NS)              // EQ_NS x EQ_NS = A^(32L)
#define EG_OFF     (P32_OFF + EQ_NS * EQ_NS + 32)       // B x NG x NS group aggregates
#define SG_OFF     (EG_OFF + EQ_B * EQ_NG * EQ_NS)      // B x NG x NS group starts
#define E_OFF      (SG_OFF + EQ_B * EQ_NG * EQ_NS)      // B x NC x NS chunk end-states
#define S_OFF      (E_OFF + EQ_B * EQ_NC * EQ_NS)       // B x NC x NS chunk start-states

typedef __attribute__((ext_vector_type(2))) float v2f;
typedef __attribute__((ext_vector_type(4))) float v4f;
typedef __attribute__((ext_vector_type(8))) float v8f;
typedef int eq_v4i __attribute__((vector_size(16)));    // matches builtin proto

#if defined(__gfx1250__) && __has_builtin(__builtin_amdgcn_global_load_async_to_lds_b128)
#define EQ_HAS_ASYNC 1
#else
#define EQ_HAS_ASYNC 0
#endif

// ---------------------------------------------------------------------------
// Pass 0: coefficients (lanes 0..9), basis sims (lanes 0..19), P^32 squarings
// ---------------------------------------------------------------------------
__global__ void eq_prep_kernel(const float* __restrict__ fc,
                               const float* __restrict__ g,
                               const float* __restrict__ q,
                               float* __restrict__ W) {
    __shared__ float sc[EQ_NB][5];
    __shared__ float Ma[EQ_NS][EQ_NS];
    __shared__ float Mb[EQ_NS][EQ_NS];
    const int lane = threadIdx.x;

    if (lane < EQ_NB) {
        const float A     = powf(10.0f, g[lane] * 0.025f);   // 10^(g/40)
        const float w0    = 2.0f * EQ_PI * fc[lane] / EQ_FS;
        const float c     = cosf(w0);
        const float s     = sinf(w0);
        const float alpha = s / (2.0f * q[lane]);
        float b0, b1, b2, a0, a1, a2;
        if (lane == 0) {                 // low shelf
            const float sqA = sqrtf(A);
            b0 = A * (A + 1.0f - (A - 1.0f) * c + 2.0f * sqA * alpha);
            b1 = 2.0f * A * (A - 1.0f - (A + 1.0f) * c);
            b2 = A * (A + 1.0f - (A - 1.0f) * c - 2.0f * sqA * alpha);
            a0 = A + 1.0f + (A - 1.0f) * c + 2.0f * sqA * alpha;
            a1 = -2.0f * (A - 1.0f + (A + 1.0f) * c);
            a2 = A + 1.0f + (A - 1.0f) * c - 2.0f * sqA * alpha;
        } else if (lane == EQ_NB - 1) {  // high shelf
            const float sqA = sqrtf(A);
            b0 = A * (A + 1.0f + (A - 1.0f) * c + 2.0f * sqA * alpha);
            b1 = -2.0f * A * (A - 1.0f + (A + 1.0f) * c);
            b2 = A * (A + 1.0f + (A - 1.0f) * c - 2.0f * sqA * alpha);
            a0 = A + 1.0f - (A - 1.0f) * c + 2.0f * sqA * alpha;
            a1 = 2.0f * (A - 1.0f - (A + 1.0f) * c);
            a2 = A + 1.0f - (A - 1.0f) * c - 2.0f * sqA * alpha;
        } else {                          // peaking
            b0 = 1.0f + alpha * A;
            b1 = -2.0f * c;
            b2 = 1.0f - alpha * A;
            a0 = 1.0f + alpha / A;
            a1 = -2.0f * c;
            a2 = 1.0f - alpha / A;
        }
        const float inv = 1.0f / a0;
        sc[lane][0] = b0 * inv; sc[lane][1] = b1 * inv; sc[lane][2] = b2 * inv;
        sc[lane][3] = a1 * inv; sc[lane][4] = a2 * inv;
        #pragma unroll
        for (int j = 0; j < 5; ++j) W[COEFF_OFF + lane * 5 + j] = sc[lane][j];
    }
    __syncthreads();

    // Basis simulations: lane i runs the exact cascade from s0 = e_i, x == 0.
    // Records Theta[t][i] (homogeneous output response) and P[:,i] = A^L e_i.
    if (lane < EQ_NS) {
        float cb0[EQ_NB], cb1[EQ_NB], cb2[EQ_NB], ca1[EQ_NB], ca2[EQ_NB];
        #pragma unroll
        for (int k = 0; k < EQ_NB; ++k) {
            cb0[k] = sc[k][0]; cb1[k] = sc[k][1]; cb2[k] = sc[k][2];
            ca1[k] = sc[k][3]; ca2[k] = sc[k][4];
        }
        float st[EQ_NS];
        #pragma unroll
        for (int i = 0; i < EQ_NS; ++i) st[i] = 0.0f;
        st[lane] = 1.0f;

        for (int t = 0; t < EQ_L; ++t) {
            float u = 0.0f;  // homogeneous: zero input
            #pragma unroll
            for (int k = 0; k < EQ_NB; ++k) {
                const float y = cb0[k] * u + st[2 * k];
                st[2 * k]     = cb1[k] * u - ca1[k] * y + st[2 * k + 1];
                st[2 * k + 1] = cb2[k] * u - ca2[k] * y;
                u = y;
            }
            W[THETA_OFF + t * EQ_NS + lane] = u;
        }
        #pragma unroll
        for (int i = 0; i < EQ_NS; ++i) {
            W[P_OFF + i * EQ_NS + lane] = st[i];   // P[i][j], column j = lane
            Ma[i][lane] = st[i];
        }
    }
    __syncthreads();

    // P32 = P^32 via 5 squarings (lane j owns column j).
    for (int itn = 0; itn < 5; ++itn) {
        if (lane < EQ_NS) {
            float col[EQ_NS];
            #pragma unroll
            for (int i = 0; i < EQ_NS; ++i) {
                float acc = 0.0f;
                #pragma unroll
                for (int k = 0; k < EQ_NS; ++k) acc += Ma[i][k] * Ma[k][lane];
                col[i] = acc;
            }
            #pragma unroll
            for (int i = 0; i < EQ_NS; ++i) Mb[i][lane] = col[i];
        }
        __syncthreads();
        if (lane < EQ_NS) {
            #pragma unroll
            for (int i = 0; i < EQ_NS; ++i) Ma[i][lane] = Mb[i][lane];
        }
        __syncthreads();
    }
    if (lane < EQ_NS) {
        #pragma unroll
        for (int i = 0; i < EQ_NS; ++i)
            W[P32_OFF + i * EQ_NS + lane] = Ma[i][lane];
    }
}

// ---------------------------------------------------------------------------
// Pass 1: zero-state chunk recurrence, LDS-staged for coalesced global I/O.
// Block = 64 threads = 64 chunks of one row segment; grid = 64 rows x 8 segs.
// ---------------------------------------------------------------------------
__global__ void eq_pass1_kernel(const float* __restrict__ x,
                                float* __restrict__ y,
                                float* __restrict__ W) {
    __shared__ float tile[P1_THREADS * P1_STRIDE];   // ~33.8 KB of 320 KB/WGP
    const int tid = threadIdx.x;
    const int row = blockIdx.x >> 3;          // 8 segments per row
    const int seg = blockIdx.x & 7;
    const size_t gbase = (size_t)row * EQ_T + (size_t)seg * (P1_THREADS * EQ_L);
    const int chunk_g = seg * P1_THREADS + tid;   // global chunk id in row

    // --- stage input into LDS, coalesced B128 (async path on gfx1250) ---
    const float* __restrict__ gsrc = x + gbase;
    #pragma unroll 4
    for (int i = 0; i < 32; ++i) {
        const int v  = i * P1_THREADS + tid;      // float4 index in segment
        const int ch = v >> 5;                    // 32 float4 per chunk
        const int tv = v & 31;
        float* dst = &tile[ch * P1_STRIDE + tv * 4];
#if EQ_HAS_ASYNC
        __builtin_amdgcn_global_load_async_to_lds_b128(
            (__attribute__((address_space(1))) eq_v4i*)(const void*)(gsrc + (size_t)v * 4),
            (__attribute__((address_space(3))) eq_v4i*)(void*)dst,
            0, 0);
#else
        *(v4f*)dst = ((const v4f*)gsrc)[v];
#endif
    }
#if EQ_HAS_ASYNC
#if __has_builtin(__builtin_amdgcn_s_wait_asynccnt)
    __builtin_amdgcn_s_wait_asynccnt(0);
#else
    asm volatile("s_wait_asynccnt 0" ::: "memory");
#endif
#endif
    __syncthreads();

    // --- per-thread exact cascade over its chunk, in-place in LDS ---
    float cb0[EQ_NB], cb1[EQ_NB], cb2[EQ_NB], ca1[EQ_NB], ca2[EQ_NB];
    #pragma unroll
    for (int k = 0; k < EQ_NB; ++k) {
        cb0[k] = W[COEFF_OFF + k * 5 + 0];
        cb1[k] = W[COEFF_OFF + k * 5 + 1];
        cb2[k] = W[COEFF_OFF + k * 5 + 2];
        ca1[k] = W[COEFF_OFF + k * 5 + 3];
        ca2[k] = W[COEFF_OFF + k * 5 + 4];
    }
    float st[EQ_NS];
    #pragma unroll
    for (int i = 0; i < EQ_NS; ++i) st[i] = 0.0f;

    float* __restrict__ lx = &tile[tid * P1_STRIDE];
    for (int t = 0; t < EQ_L; ++t) {
        float u = lx[t];
        #pragma unroll
        for (int k = 0; k < EQ_NB; ++k) {
            const float yy = cb0[k] * u + st[2 * k];
            st[2 * k]     = cb1[k] * u - ca1[k] * yy + st[2 * k + 1];
            st[2 * k + 1] = cb2[k] * u - ca2[k] * yy;
            u = yy;
        }
        lx[t] = u;                                  // overwrite x with y
    }
    float* __restrict__ e = W + E_OFF + ((size_t)row * EQ_NC + chunk_g) * EQ_NS;
    #pragma unroll
    for (int i = 0; i < EQ_NS; ++i) e[i] = st[i];
    __syncthreads();

    // --- write provisional outputs back, coalesced B128 ---
    v4f* __restrict__ gdst = (v4f*)(y + gbase);
    #pragma unroll 4
    for (int i = 0; i < 32; ++i) {
        const int v  = i * P1_THREADS + tid;
        const int ch = v >> 5;
        const int tv = v & 31;
        gdst[v] = *(const v4f*)&tile[ch * P1_STRIDE + tv * 4];
    }
}

// ---------------------------------------------------------------------------
// Pass 2: hierarchical scan. Lane i carries state component i; matrix rows in
// registers; cross-lane broadcast via wave32 shuffles.
// ---------------------------------------------------------------------------
__device__ __forceinline__ float eq_matvec_step(const float pr[EQ_NS], float s,
                                                float e) {
    float a0 = e, a1 = 0.0f;
    #pragma unroll
    for (int j = 0; j < EQ_NS; j += 2) {
        a0 += pr[j]     * __shfl(s, j, 32);
        a1 += pr[j + 1] * __shfl(s, j + 1, 32);
    }
    return a0 + a1;
}

// (a) per-(row,group): aggregate 32 chunks from zero -> Eg
__global__ void eq_scan_a_kernel(float* __restrict__ W) {
    const int row = blockIdx.x >> 4, grp = blockIdx.x & 15, lane = threadIdx.x;
    float pr[EQ_NS];
    #pragma unroll
    for (int j = 0; j < EQ_NS; ++j)
        pr[j] = (lane < EQ_NS) ? W[P_OFF + lane * EQ_NS + j] : 0.0f;
    float s = 0.0f;
    for (int c = grp * EQ_GRP; c < (grp + 1) * EQ_GRP; ++c) {
        const float e = (lane < EQ_NS)
            ? W[E_OFF + ((size_t)row * EQ_NC + c) * EQ_NS + lane] : 0.0f;
        s = eq_matvec_step(pr, s, e);
    }
    if (lane < EQ_NS)
        W[EG_OFF + ((size_t)row * EQ_NG + grp) * EQ_NS + lane] = s;
}

// (b) per-row: scan 16 groups with P32 -> group start states Sg
__global__ void eq_scan_b_kernel(float* __restrict__ W) {
    const int row = blockIdx.x, lane = threadIdx.x;
    float pr[EQ_NS];
    #pragma unroll
    for (int j = 0; j < EQ_NS; ++j)
        pr[j] = (lane < EQ_NS) ? W[P32_OFF + lane * EQ_NS + j] : 0.0f;
    float s = 0.0f;
    for (int gI = 0; gI < EQ_NG; ++gI) {
        const size_t idx = ((size_t)row * EQ_NG + gI) * EQ_NS + lane;
        if (lane < EQ_NS) W[SG_OFF + idx] = s;
        const float e = (lane < EQ_NS) ? W[EG_OFF + idx] : 0.0f;
        s = eq_matvec_step(pr, s, e);
    }
}

// (c) per-(row,group): re-scan 32 chunks from true group start -> S per chunk
__global__ void eq_scan_c_kernel(float* __restrict__ W) {
    const int row = blockIdx.x >> 4, grp = blockIdx.x & 15, lane = threadIdx.x;
    float pr[EQ_NS];
    #pragma unroll
    for (int j = 0; j < EQ_NS; ++j)
        pr[j] = (lane < EQ_NS) ? W[P_OFF + lane * EQ_NS + j] : 0.0f;
    float s = (lane < EQ_NS)
        ? W[SG_OFF + ((size_t)row * EQ_NG + grp) * EQ_NS + lane] : 0.0f;
    for (int c = grp * EQ_GRP; c < (grp + 1) * EQ_GRP; ++c) {
        const size_t idx = ((size_t)row * EQ_NC + c) * EQ_NS + lane;
        if (lane < EQ_NS) W[S_OFF + idx] = s;
        const float e = (lane < EQ_NS) ? W[E_OFF + idx] : 0.0f;
        s = eq_matvec_step(pr, s, e);
    }
}

// ---------------------------------------------------------------------------
// Pass 3: y += Theta (L x 20) * S (20 x 64) per chunk, f32 WMMA 16x16x4.
// One wave per 16x16 output tile; K = 20 = 5 wmma k-steps.
// ---------------------------------------------------------------------------
__global__ void eq_fixup_kernel(float* __restrict__ y,
                                const float* __restrict__ W) {
    const int lane = threadIdx.x & 31;
    const int wave = threadIdx.x >> 5;
    const int job  = blockIdx.x * 8 + wave;   // NC * 8 mtiles * 4 ntiles = 16384
    const int chunk = job >> 5;               // 32 tiles per chunk
    const int rem   = job & 31;
    const int mt    = rem >> 2;               // time tile within chunk (0..7)
    const int nt    = rem & 3;                // batch tile (0..3)

    const float* __restrict__ Th = W + THETA_OFF;
    const float* __restrict__ S  = W + S_OFF;

    const int half = (lane < 16) ? 0 : 1;
    const int l15  = lane & 15;

    v8f acc = {0.f, 0.f, 0.f, 0.f, 0.f, 0.f, 0.f, 0.f};

    #pragma unroll
    for (int kk = 0; kk < 5; ++kk) {
        const int kb = kk * 4 + 2 * half;     // this half-wave's K pair
        v2f a, b;
        const int m = mt * 16 + l15;          // time row within chunk
        a[0] = Th[m * EQ_NS + kb];
        a[1] = Th[m * EQ_NS + kb + 1];
        const int n = nt * 16 + l15;          // batch row
        const float* __restrict__ sp = S + ((size_t)n * EQ_NC + chunk) * EQ_NS;
        b[0] = sp[kb];
        b[1] = sp[kb + 1];
        acc = __builtin_amdgcn_wmma_f32_16x16x4_f32(
            false, a, false, b, (short)0, acc, false, false);
    }

    const int r = nt * 16 + l15;
    const size_t base = (size_t)r * EQ_T + (size_t)chunk * EQ_L + mt * 16;
    #pragma unroll
    for (int v = 0; v < 8; ++v) {
        const int tl = v + 8 * half;          // D layout: rows v / v+8
        y[base + tl] += acc[v];
    }
}

// ---------------------------------------------------------------------------
extern "C" void kernel_launch(void* const* d_in, const int* in_sizes, int n_in,
                              void* d_out, int out_size, void* d_ws, size_t ws_size,
                              hipStream_t stream) {
    (void)in_sizes; (void)n_in; (void)out_size; (void)ws_size;
    const float* audio = (const float*)d_in[0];
    const float* fc    = (const float*)d_in[1];
    const float* g     = (const float*)d_in[2];
    const float* q     = (const float*)d_in[3];
    float* out = (float*)d_out;
    float* W   = (float*)d_ws;

    eq_prep_kernel  <<<1, 32, 0, stream>>>(fc, g, q, W);
    eq_pass1_kernel <<<EQ_B * P1_SEGS, P1_THREADS, 0, stream>>>(audio, out, W);
    eq_scan_a_kernel<<<EQ_B * EQ_NG, 32, 0, stream>>>(W);
    eq_scan_b_kernel<<<EQ_B, 32, 0, stream>>>(W);
    eq_scan_c_kernel<<<EQ_B * EQ_NG, 32, 0, stream>>>(W);
    eq_fixup_kernel <<<(EQ_NC * 32) / 8, 256, 0, stream>>>(out, W);
}